// Voxelization_63960652972185
// MI455X (gfx1250) — compile-verified
//
#include <hip/hip_runtime.h>
#include <stdint.h>

// Problem constants (from reference setup_inputs / RES)
#define RES   32
#define RES3  (RES * RES * RES)          // 32768
#define NB    8
#define NC    64
#define NPT   100000
#define GRID_ELEMS (NB * NC * RES3)      // 16,777,216 floats (64 MB)

// Workspace layout (floats)
#define WS_COUNTS 0                       // NB*RES3 counts
#define WS_SUMS   (NB * RES3)             // NB*3 coordinate sums
#define WS_MAXR2  (NB * RES3 + NB * 3)    // NB max r^2 (uint-bits atomicMax)
#define WS_TOTAL  (NB * RES3 + NB * 3 + NB)  // 262,176 floats

// LDS layout inside k_scatter (floats): [0,16384) feature tile 64x256,
// [16384, 16384+3*256) coords tile 3xtile_w (TDM destination)
#define LDS_FEAT_FLOATS (NC * 256)            // 16384
#define LDS_COORD_FLOATS (3 * 256)            // 768
#define LDS_TOTAL_BYTES ((LDS_FEAT_FLOATS + LDS_COORD_FLOATS) * 4)

typedef uint32_t u32x4 __attribute__((ext_vector_type(4)));
typedef uint32_t u32x8 __attribute__((ext_vector_type(8)));

__device__ __forceinline__ void atomic_add_f32(float* p, float v) {
  // relaxed agent-scope fp32 atomic -> global_atomic_add_f32 (no return)
  __hip_atomic_fetch_add(p, v, __ATOMIC_RELAXED, __HIP_MEMORY_SCOPE_AGENT);
}

// One-shot TDM: 2D tile (tile_dim1 rows x tile_dim0 elems, row stride
// tensor_dim0_stride) from global -> LDS. Descriptor per CDNA5 ISA ch.8.
__device__ __forceinline__ void tdm_load_2d(uint32_t lds_byte_off,
                                            const void* gptr,
                                            uint32_t tensor_d0,
                                            uint32_t tensor_d1,
                                            uint32_t tile_d0,
                                            uint32_t tile_d1,
                                            uint32_t d0_stride_elems) {
  const uint64_t ga = (uint64_t)(uintptr_t)gptr;
  u32x4 g0;
  g0[0] = 1u;                                        // count=1 (valid user D#)
  g0[1] = lds_byte_off;                              // lds_addr
  g0[2] = (uint32_t)ga;                              // global_addr[31:0]
  g0[3] = (uint32_t)((ga >> 32) & 0x01FFFFFFu)       // global_addr[56:32]
        | (2u << 30);                                // type=2 ("image")
  u32x8 g1;
  g1[0] = 0x00020000u;                               // data_size=2 (4 bytes)
  g1[1] = (tensor_d0 & 0xFFFFu) << 16;               // tensor_dim0[15:0]
  g1[2] = (tensor_d0 >> 16) | ((tensor_d1 & 0xFFFFu) << 16); // dim0 hi | dim1 lo
  g1[3] = (tensor_d1 >> 16) | (tile_d0 << 16);       // dim1 hi | tile_dim0
  g1[4] = tile_d1;                                   // tile_dim1 (tile_dim2=0)
  g1[5] = d0_stride_elems;                           // tensor_dim0_stride[31:0]
  g1[6] = 0u;                                        // stride hi / dim1_stride lo
  g1[7] = 0u;
  asm volatile("tensor_load_to_lds %0, %1" :: "s"(g0), "s"(g1) : "memory");
}

__global__ void k_zero4(float4* __restrict__ p, int n4) {
  int i = blockIdx.x * blockDim.x + threadIdx.x;
  const int stride = gridDim.x * blockDim.x;
  const float4 z = make_float4(0.f, 0.f, 0.f, 0.f);
  for (; i < n4; i += stride) p[i] = z;
}

// Per-(batch,dim) coordinate sums: block tree-reduce + one atomic per block.
__global__ void __launch_bounds__(256)
k_mean(const float* __restrict__ coords, float* __restrict__ sums) {
  __shared__ float red[256];
  const int row = blockIdx.y;                        // b*3 + dim, 0..23
  const float* src = coords + (size_t)row * NPT;
  float s = 0.f;
  for (int i = blockIdx.x * blockDim.x + threadIdx.x; i < NPT;
       i += gridDim.x * blockDim.x)
    s += src[i];
  red[threadIdx.x] = s;
  __syncthreads();
  for (int off = 128; off > 0; off >>= 1) {
    if (threadIdx.x < (unsigned)off) red[threadIdx.x] += red[threadIdx.x + off];
    __syncthreads();
  }
  if (threadIdx.x == 0) atomic_add_f32(&sums[row], red[0]);
}

// Per-batch max squared radius. r^2 >= 0 so float order == uint order.
__global__ void __launch_bounds__(256)
k_radius(const float* __restrict__ coords, const float* __restrict__ sums,
         unsigned* __restrict__ maxr2) {
  __shared__ float red[256];
  const int b = blockIdx.y;
  const float mx = sums[b * 3 + 0] * (1.0f / NPT);
  const float my = sums[b * 3 + 1] * (1.0f / NPT);
  const float mz = sums[b * 3 + 2] * (1.0f / NPT);
  const float* cb = coords + (size_t)b * 3 * NPT;
  float m = 0.f;
  for (int i = blockIdx.x * blockDim.x + threadIdx.x; i < NPT;
       i += gridDim.x * blockDim.x) {
    const float dx = cb[i] - mx;
    const float dy = cb[NPT + i] - my;
    const float dz = cb[2 * NPT + i] - mz;
    m = fmaxf(m, dx * dx + dy * dy + dz * dz);
  }
  red[threadIdx.x] = m;
  __syncthreads();
  for (int off = 128; off > 0; off >>= 1) {
    if (threadIdx.x < (unsigned)off)
      red[threadIdx.x] = fmaxf(red[threadIdx.x], red[threadIdx.x + off]);
    __syncthreads();
  }
  if (threadIdx.x == 0) atomicMax(&maxr2[b], __float_as_uint(red[0]));
}

// Fused scatter:
//  - per-lane async B128 stages the 64x256 feature tile (ASYNCcnt path)
//  - one TDM descriptor DMAs the 3 x tile_w coords tile (TENSORcnt path)
//  - coord math / norm_coords / count atomics overlap the feature stream
//  - drain LDS into the L2-resident fp32 atomic accumulators
__global__ void __launch_bounds__(256)
k_scatter(const float* __restrict__ feats, const float* __restrict__ coords,
          const float* __restrict__ sums, const unsigned* __restrict__ maxr2,
          float* __restrict__ grid_sums, float* __restrict__ counts,
          float* __restrict__ norm_out) {
  extern __shared__ float smem[];
  const int tid  = threadIdx.x;
  const int b    = blockIdx.y;
  const int n0   = blockIdx.x << 8;                  // tile base point
  const int n    = n0 + tid;
  const int nrem = NPT - n0;
  const int tile_w = nrem < 256 ? nrem : 256;

  // LDS aperture keeps the byte offset in the low 32 bits of the flat addr.
  const unsigned lds_base = (unsigned)(uintptr_t)smem;
  const float* gbase = feats + (size_t)b * NC * NPT + n0;

  // 4096 x b128 feature tile loads over 256 threads -> 16 async ops/thread.
#pragma unroll
  for (int i = 0; i < 16; ++i) {
    const int lin = (i << 8) + tid;                  // 0..4095
    const int c   = lin >> 6;                        // channel 0..63
    const int pg  = (lin & 63) << 2;                 // point group *4
    if (pg < nrem) {
      const unsigned loff = lds_base + (unsigned)((c << 8) + pg) * 4u;
      const float* gp = gbase + (size_t)c * NPT + pg;
      asm volatile("global_load_async_to_lds_b128 %0, %1, off"
                   :: "v"(loff), "v"(gp)
                   : "memory");
    }
  }

  // Wave 0 drives the TDM for the coords tile (TDM ignores EXEC; issue once).
  if (tid < 32) {
    tdm_load_2d(lds_base + LDS_FEAT_FLOATS * 4u,
                coords + (size_t)b * 3 * NPT + n0,
                (uint32_t)nrem, 3u,                  // tensor dims (OOB guard)
                (uint32_t)tile_w, 3u,                // tile dims
                (uint32_t)NPT);                      // row stride
    __builtin_amdgcn_s_wait_tensorcnt(0);
  }
  __syncthreads();                                   // coords visible in LDS

  int v = 0;
  const bool valid = (n < NPT);
  if (valid) {
    const float inv = 1.0f / (2.0f * sqrtf(__uint_as_float(maxr2[b])));
    const float mx = sums[b * 3 + 0] * (1.0f / NPT);
    const float my = sums[b * 3 + 1] * (1.0f / NPT);
    const float mz = sums[b * 3 + 2] * (1.0f / NPT);

    const float* csh = smem + LDS_FEAT_FLOATS;       // 3 rows x tile_w
    const float x = csh[tid];
    const float y = csh[tile_w + tid];
    const float z = csh[2 * tile_w + tid];

    float fx = ((x - mx) * inv + 0.5f) * (float)RES;
    float fy = ((y - my) * inv + 0.5f) * (float)RES;
    float fz = ((z - mz) * inv + 0.5f) * (float)RES;
    fx = fminf(fmaxf(fx, 0.f), (float)(RES - 1));
    fy = fminf(fmaxf(fy, 0.f), (float)(RES - 1));
    fz = fminf(fmaxf(fz, 0.f), (float)(RES - 1));

    float* nb_ = norm_out + (size_t)b * 3 * NPT;
    nb_[n]           = fx;
    nb_[NPT + n]     = fy;
    nb_[2 * NPT + n] = fz;

    const int vx = (int)rintf(fx);                   // RNE == jnp.round
    const int vy = (int)rintf(fy);
    const int vz = (int)rintf(fz);
    v = (vx * RES + vy) * RES + vz;

    atomic_add_f32(&counts[(b << 15) + v], 1.0f);
  }

  // Drain: own-wave async completion, then cross-wave visibility.
  asm volatile("s_wait_asynccnt 0x0" ::: "memory");
  __syncthreads();

  if (valid) {
    float* gb2 = grid_sums + ((size_t)(b * NC) << 15) + v;
#pragma unroll 8
    for (int c = 0; c < NC; ++c)
      atomic_add_f32(gb2 + ((size_t)c << 15), smem[(c << 8) + tid]);
  }
}

// In-place average: grid[b,c,v] /= max(count[b,v], 1)
__global__ void __launch_bounds__(256)
k_div(float* __restrict__ grid, const float* __restrict__ counts) {
  const int i = blockIdx.x * blockDim.x + threadIdx.x;   // 0..GRID_ELEMS-1
  const int v = i & (RES3 - 1);
  const int b = i >> 21;                                 // NC*RES3 = 2^21
  grid[i] = grid[i] / fmaxf(counts[(b << 15) + v], 1.0f);
}

extern "C" void kernel_launch(void* const* d_in, const int* in_sizes, int n_in,
                              void* d_out, int out_size, void* d_ws, size_t ws_size,
                              hipStream_t stream) {
  (void)in_sizes; (void)n_in; (void)out_size; (void)ws_size;
  const float* feats  = (const float*)d_in[0];   // [8,64,100000]
  const float* coords = (const float*)d_in[1];   // [8,3,100000]

  float* out      = (float*)d_out;
  float* grid     = out;                         // [8,64,32,32,32]
  float* norm_out = out + GRID_ELEMS;            // [8,3,100000]

  float*    ws     = (float*)d_ws;
  float*    counts = ws + WS_COUNTS;
  float*    sums   = ws + WS_SUMS;
  unsigned* maxr2  = (unsigned*)(ws + WS_MAXR2);

  // Zero accumulators every call (harness poisons once; we accumulate).
  k_zero4<<<dim3(4096), 256, 0, stream>>>((float4*)grid, GRID_ELEMS / 4);
  k_zero4<<<dim3(256),  256, 0, stream>>>((float4*)ws,   WS_TOTAL / 4);

  k_mean  <<<dim3(40, NB * 3), 256, 0, stream>>>(coords, sums);
  k_radius<<<dim3(64, NB),     256, 0, stream>>>(coords, sums, maxr2);

  const int tiles = (NPT + 255) / 256;           // 391
  k_scatter<<<dim3(tiles, NB), 256, LDS_TOTAL_BYTES, stream>>>(
      feats, coords, sums, maxr2, grid, counts, norm_out);

  k_div<<<dim3(GRID_ELEMS / 256), 256, 0, stream>>>(grid, counts);
}